// GNN_30399778521510
// MI455X (gfx1250) — compile-verified
//
#include <hip/hip_runtime.h>
#include <math.h>

#define HEADS 4
#define KDIM 128
#define LDS_STRIDE 132   // 128 + 4 pad -> lane l16 hits bank (l16*4)%64, conflict-free

typedef __attribute__((ext_vector_type(2))) float v2f;
typedef __attribute__((ext_vector_type(8))) float v8f;

// ---------------------------------------------------------------------------
// Y[M x OUTC] = X[M x 128] @ W[128 x OUTC], all f32, via V_WMMA_F32_16X16X4_F32
// OUTC is a template constant so every W-load / Y-store folds into
// base + 24-bit immediate offset; K-loop fully unrolled -> 32 wmma per wave.
// grid.x = M/16 row tiles, grid.y = OUTC/128 col super-tiles, 256 thr = 8 waves
// ---------------------------------------------------------------------------
template <int OUTC>
__global__ __launch_bounds__(256)
void gat_gemm_wmma(const float* __restrict__ X, const float* __restrict__ W,
                   float* __restrict__ Y) {
  __shared__ float Xs[16 * LDS_STRIDE];
  const int tid = threadIdx.x;
  const int rowBase = blockIdx.x * 16;

  // cooperative, coalesced load of the 16x128 X-tile into padded LDS
  for (int i = tid; i < 16 * (KDIM / 4); i += 256) {
    int r  = i >> 5;            // / 32
    int c4 = (i & 31) << 2;
    float4 v = *(const float4*)(X + (size_t)(rowBase + r) * KDIM + c4);
    *(float4*)(&Xs[r * LDS_STRIDE + c4]) = v;
  }
  __syncthreads();

  const int wave = tid >> 5;
  const int lane = tid & 31;
  const int l16  = lane & 15;
  const int hi   = lane >> 4;                 // half-wave select
  const int colBase = blockIdx.y * 128 + wave * 16;

  v8f c = {0.f, 0.f, 0.f, 0.f, 0.f, 0.f, 0.f, 0.f};
  const float* xrow = &Xs[l16 * LDS_STRIDE + 2 * hi];      // A: K = v + 2*hi
  const float* wp   = W + (size_t)(2 * hi) * OUTC + colBase + l16;  // B: K = v + 2*hi

#pragma unroll
  for (int kk = 0; kk < KDIM; kk += 4) {
    v2f a; a.x = xrow[kk];            a.y = xrow[kk + 1];
    v2f b; b.x = wp[kk * OUTC];       b.y = wp[(kk + 1) * OUTC];
    c = __builtin_amdgcn_wmma_f32_16x16x4_f32(false, a, false, b, (short)0, c,
                                              false, false);
  }

  // C/D layout: row = v + 8*hi, col = l16
  float* out = Y + ((size_t)rowBase + 8 * hi) * OUTC + colBase + l16;
#pragma unroll
  for (int v = 0; v < 8; ++v)
    out[v * OUTC] = c[v];
}

// ---------------------------------------------------------------------------
// attention logits: el[n,h] = feat[n,h,:].al[h,:], er likewise
// ---------------------------------------------------------------------------
__global__ void eler_kernel(const float* __restrict__ feat, const float* __restrict__ al,
                            const float* __restrict__ ar, float* __restrict__ el,
                            float* __restrict__ er, int NH, int D) {
  int i = blockIdx.x * blockDim.x + threadIdx.x;
  if (i >= NH) return;
  int n = i >> 2, hh = i & 3;
  const float* f = feat + (size_t)n * HEADS * D + hh * D;
  const float* a = al + hh * D;
  const float* b = ar + hh * D;
  float sl = 0.f, sr = 0.f;
  for (int d = 0; d < D; ++d) { float fv = f[d]; sl += fv * a[d]; sr += fv * b[d]; }
  el[i] = sl; er[i] = sr;
}

__device__ __forceinline__ float leaky02(float v) { return v >= 0.f ? v : 0.2f * v; }
// monotone float<->uint encoding so atomicMax(u32) == max(float)
__device__ __forceinline__ unsigned fenc(float v) {
  unsigned b = __float_as_uint(v);
  return (b & 0x80000000u) ? ~b : (b | 0x80000000u);
}
__device__ __forceinline__ float fdec(unsigned u) {
  return (u & 0x80000000u) ? __uint_as_float(u ^ 0x80000000u) : __uint_as_float(~u);
}

__global__ void init_kernel(unsigned* __restrict__ mEnc, float* __restrict__ denom, int NH) {
  int i = blockIdx.x * blockDim.x + threadIdx.x;
  if (i >= NH) return;
  mEnc[i] = 0u;            // below fenc(any finite) and fenc(-inf)
  denom[i] = 0.f;
}

__global__ void edge_max_kernel(const float* __restrict__ el, const float* __restrict__ er,
                                const int* __restrict__ src, const int* __restrict__ dst,
                                unsigned* __restrict__ mEnc, int EH) {
  int i = blockIdx.x * blockDim.x + threadIdx.x;
  if (i >= EH) return;
  int e = i >> 2, hh = i & 3;
  int s = src[e], d = dst[e];
  float v = leaky02(el[s * HEADS + hh] + er[d * HEADS + hh]);
  atomicMax(&mEnc[d * HEADS + hh], fenc(v));
}

__global__ void edge_exp_kernel(const float* __restrict__ el, const float* __restrict__ er,
                                const int* __restrict__ src, const int* __restrict__ dst,
                                const unsigned* __restrict__ mEnc, float* __restrict__ denom,
                                float* __restrict__ exArr, int EH) {
  int i = blockIdx.x * blockDim.x + threadIdx.x;
  if (i >= EH) return;
  int e = i >> 2, hh = i & 3;
  int s = src[e], d = dst[e];
  float v = leaky02(el[s * HEADS + hh] + er[d * HEADS + hh]);
  float m = fdec(mEnc[d * HEADS + hh]);
  float ex = __expf(v - m);
  exArr[i] = ex;
  unsafeAtomicAdd(&denom[d * HEADS + hh], ex);
}

// ---------------------------------------------------------------------------
// rst[dst] += feat[src] * alpha  — one wave32 per edge, float4 lanes, f32 atomics
// ---------------------------------------------------------------------------
__global__ void edge_agg_kernel(const float* __restrict__ feat, const float* __restrict__ exArr,
                                const float* __restrict__ denom, const int* __restrict__ src,
                                const int* __restrict__ dst, float* __restrict__ rst,
                                int E, int D) {
  int e = blockIdx.x * 8 + (threadIdx.x >> 5);
  if (e >= E) return;
  int lane = threadIdx.x & 31;
  int HD = HEADS * D;
  int PE = HD >> 5;                    // floats per lane: 4 (D=32) or 8 (D=64)
  int s = src[e], d = dst[e];
  const float* fs = feat + (size_t)s * HD;
  float* rd = rst + (size_t)d * HD;
  for (int j = 0; j < PE; j += 4) {
    int idx = lane * PE + j;           // stays within a single head (4|32, 8|64)
    int hh = idx / D;
    float alpha = exArr[(size_t)e * HEADS + hh] / denom[(size_t)d * HEADS + hh];
    float4 fv = *(const float4*)(fs + idx);
    unsafeAtomicAdd(rd + idx + 0, fv.x * alpha);
    unsafeAtomicAdd(rd + idx + 1, fv.y * alpha);
    unsafeAtomicAdd(rd + idx + 2, fv.z * alpha);
    unsafeAtomicAdd(rd + idx + 3, fv.w * alpha);
  }
}

__global__ void copy_kernel(const float* __restrict__ a, float* __restrict__ b, int n) {
  int i = blockIdx.x * blockDim.x + threadIdx.x;
  if (i < n) b[i] = a[i];
}

__global__ void relu_kernel(const float* __restrict__ a, float* __restrict__ b, int n) {
  int i = blockIdx.x * blockDim.x + threadIdx.x;
  if (i < n) b[i] = fmaxf(a[i], 0.f);
}

__global__ void head_mean_kernel(const float* __restrict__ x, float* __restrict__ out, int N) {
  int i = blockIdx.x * blockDim.x + threadIdx.x;
  if (i >= N * 64) return;
  int n = i >> 6, d = i & 63;
  const float* p = x + (size_t)n * 256;
  out[i] = 0.25f * (p[d] + p[64 + d] + p[128 + d] + p[192 + d]);
}

// ---------------------------------------------------------------------------
extern "C" void kernel_launch(void* const* d_in, const int* in_sizes, int n_in,
                              void* d_out, int out_size, void* d_ws, size_t ws_size,
                              hipStream_t stream) {
  const float* h     = (const float*)d_in[0];
  const float* W1    = (const float*)d_in[1];
  const float* al1   = (const float*)d_in[2];
  const float* ar1   = (const float*)d_in[3];
  const float* W2    = (const float*)d_in[4];
  const float* al2   = (const float*)d_in[5];
  const float* ar2   = (const float*)d_in[6];
  const float* W3    = (const float*)d_in[7];
  const float* al3   = (const float*)d_in[8];
  const float* ar3   = (const float*)d_in[9];
  const float* resW3 = (const float*)d_in[10];
  const int*   src   = (const int*)d_in[11];
  const int*   dst   = (const int*)d_in[12];
  float* out = (float*)d_out;

  const int N = in_sizes[0] / KDIM;   // 50000 (divisible by 16)
  const int E = in_sizes[11];         // 800000

  // workspace layout (all f32 unless noted)
  float*    bufF  = (float*)d_ws;                       // feat   [N*256]
  float*    bufR  = bufF + (size_t)N * 256;             // rst    [N*256]
  float*    bufX  = bufR + (size_t)N * 256;             // x      [N*128]
  float*    el    = bufX + (size_t)N * 128;             // [N*H]
  float*    er    = el + (size_t)N * HEADS;             // [N*H]
  unsigned* mEnc  = (unsigned*)(er + (size_t)N * HEADS);// [N*H]
  float*    denom = (float*)(mEnc + (size_t)N * HEADS); // [N*H]
  float*    exArr = denom + (size_t)N * HEADS;          // [E*H]

  const int NH = N * HEADS, EH = E * HEADS;

  auto gemm = [&](const float* x, const float* W, float* y, int HD) {
    dim3 gg(N / 16, HD / 128);
    if (HD == 128)
      gat_gemm_wmma<128><<<gg, 256, 0, stream>>>(x, W, y);
    else
      gat_gemm_wmma<256><<<gg, 256, 0, stream>>>(x, W, y);
  };

  auto layer = [&](const float* x, const float* W, const float* al, const float* ar,
                   int D, const float* resW, bool do_relu, float* xout) {
    const int HD = HEADS * D;
    gemm(x, W, bufF, HD);
    eler_kernel<<<(NH + 255) / 256, 256, 0, stream>>>(bufF, al, ar, el, er, NH, D);
    if (resW)  // residual projection doubles as rst initializer
      gemm(x, resW, bufR, HD);
    else
      copy_kernel<<<(N * HD + 255) / 256, 256, 0, stream>>>(x, bufR, N * HD);
    init_kernel<<<(NH + 255) / 256, 256, 0, stream>>>(mEnc, denom, NH);
    edge_max_kernel<<<(EH + 255) / 256, 256, 0, stream>>>(el, er, src, dst, mEnc, EH);
    edge_exp_kernel<<<(EH + 255) / 256, 256, 0, stream>>>(el, er, src, dst, mEnc, denom, exArr, EH);
    edge_agg_kernel<<<(E + 7) / 8, 256, 0, stream>>>(bufF, exArr, denom, src, dst, bufR, E, D);
    if (do_relu)
      relu_kernel<<<(N * HD + 255) / 256, 256, 0, stream>>>(bufR, xout, N * HD);
  };

  layer(h,    W1, al1, ar1, 32, nullptr, true,  bufX);
  layer(bufX, W2, al2, ar2, 32, nullptr, true,  bufX);
  layer(bufX, W3, al3, ar3, 64, resW3,   false, nullptr);
  head_mean_kernel<<<(N * 64 + 255) / 256, 256, 0, stream>>>(bufR, out, N);
}